// RatioEstimatorCube_21672404975686
// MI455X (gfx1250) — compile-verified
//
#include <hip/hip_runtime.h>

typedef __attribute__((ext_vector_type(2))) float v2f;
typedef __attribute__((ext_vector_type(8))) float v8f;

#define BINS_PER_BATCH 262144      // 64*64*64
#define NCOUNT 16777216            // 64 batches * 262144

// ---------------- Kernel 1: zero the counts half of d_out ----------------
__global__ __launch_bounds__(256) void zero_counts_kernel(float4* __restrict__ p) {
    size_t i = (size_t)blockIdx.x * 256 + threadIdx.x;   // 4,194,304 float4s
    p[i] = make_float4(0.f, 0.f, 0.f, 0.f);
}

// ---------------- Kernel 2: histogram via global float atomics ----------------
// counts <= 4096 so every f32 add of 1.0 is exact -> order-independent, deterministic.
__global__ __launch_bounds__(256) void hist_kernel(const float* __restrict__ z,
                                                   float* __restrict__ counts) {
    int gid = blockIdx.x * 256 + threadIdx.x;            // 0..262143 (B*P)
    int b = gid >> 12;                                   // / 4096
    float zx = z[3 * gid + 0];
    float zy = z[3 * gid + 1];
    float zz = z[3 * gid + 2];
    int ix = (int)floorf(zx * 64.0f); ix = ix < 0 ? 0 : (ix > 63 ? 63 : ix);
    int iy = (int)floorf(zy * 64.0f); iy = iy < 0 ? 0 : (iy > 63 ? 63 : iy);
    int iz = (int)floorf(zz * 64.0f); iz = iz < 0 ? 0 : (iz > 63 ? 63 : iz);
    int lin = (ix << 12) + (iy << 6) + iz;
    atomicAdd(&counts[(size_t)b * BINS_PER_BATCH + lin], 1.0f);
}

// ---------------- Kernel 3: suffix-sum along last dim via triangular WMMA + mask ----------------
// Each block: 128 rows of 64 f32 (32 KiB) staged in LDS. Each wave: 16 rows.
// O(16x64) = C(16x64) x M(64x64), M[k][j] = (k >= j)  -> reverse inclusive cumsum.
// Triangular structure: output column tile jt only receives K-tiles kt >= 4*jt.
__global__ __launch_bounds__(256) void scan_mask_kernel(const float* __restrict__ x,
                                                        float* __restrict__ counts,  // in/out
                                                        float* __restrict__ rmask) {
    __shared__ float lds[8192];                          // 128 rows * 64 f32
    const int tid = threadIdx.x;
    const size_t blockBase = (size_t)blockIdx.x * 8192;  // float index

    // Stage: coalesced 128-bit loads, 8 per thread.
    const float4* src = (const float4*)(counts + blockBase);
    float4* dst = (float4*)lds;
#pragma unroll
    for (int i = 0; i < 8; ++i)
        dst[i * 256 + tid] = src[i * 256 + tid];
    __syncthreads();

    const int wave = tid >> 5;
    const int lane = tid & 31;
    const int half = lane >> 4;      // A/B: lanes 0-15 hold K=2k,2k+1 ; lanes 16-31 hold K=2k+2,2k+3
    const int lm   = lane & 15;      // A: row M ; B/C/D: column N
    const float* rowsrc = &lds[wave * 1024];   // this wave's 16x64 block

    v8f acc0 = {}, acc1 = {}, acc2 = {}, acc3 = {};   // output column tiles jt=0..3

#pragma unroll
    for (int kt = 0; kt < 16; ++kt) {
        const int gk = kt * 4 + 2 * half;    // global K of this lane's first element
        // A fragment: A[m][k] = counts[row lm][gk], counts[row lm][gk+1]
        v2f a = *(const v2f*)&rowsrc[lm * 64 + gk];
        // B fragments: B[k][n] = (global_k >= global_j), global_j = jt*16 + lm.
        // Tile is identically zero when max(gk+1) < 16*jt  <=>  kt < 4*jt: skip those WMMAs.
        {
            v2f b;
            b.x = (gk     >=  0 + lm) ? 1.f : 0.f;
            b.y = (gk + 1 >=  0 + lm) ? 1.f : 0.f;
            acc0 = __builtin_amdgcn_wmma_f32_16x16x4_f32(false, a, false, b, (short)0, acc0, false, false);
        }
        if (kt >= 4) {
            v2f b;
            b.x = (gk     >= 16 + lm) ? 1.f : 0.f;
            b.y = (gk + 1 >= 16 + lm) ? 1.f : 0.f;
            acc1 = __builtin_amdgcn_wmma_f32_16x16x4_f32(false, a, false, b, (short)0, acc1, false, false);
        }
        if (kt >= 8) {
            v2f b;
            b.x = (gk     >= 32 + lm) ? 1.f : 0.f;
            b.y = (gk + 1 >= 32 + lm) ? 1.f : 0.f;
            acc2 = __builtin_amdgcn_wmma_f32_16x16x4_f32(false, a, false, b, (short)0, acc2, false, false);
        }
        if (kt >= 12) {
            v2f b;
            b.x = (gk     >= 48 + lm) ? 1.f : 0.f;
            b.y = (gk + 1 >= 48 + lm) ? 1.f : 0.f;
            acc3 = __builtin_amdgcn_wmma_f32_16x16x4_f32(false, a, false, b, (short)0, acc3, false, false);
        }
    }

    // C/D layout: VGPR v -> M = v + 8*half, N = lm (per ISA 16x16 f32 layout).
    const size_t waveBase = blockBase + (size_t)wave * 1024;
#pragma unroll
    for (int v = 0; v < 8; ++v) {
        const int M = v + 8 * half;
        const size_t o = waveBase + (size_t)M * 64 + lm;
        const float c0 = acc0[v], c1 = acc1[v], c2 = acc2[v], c3 = acc3[v];
        counts[o +  0] = c0;  rmask[o +  0] = (c0 > 0.f) ? x[o +  0] : 0.f;
        counts[o + 16] = c1;  rmask[o + 16] = (c1 > 0.f) ? x[o + 16] : 0.f;
        counts[o + 32] = c2;  rmask[o + 32] = (c2 > 0.f) ? x[o + 32] : 0.f;
        counts[o + 48] = c3;  rmask[o + 48] = (c3 > 0.f) ? x[o + 48] : 0.f;
    }
}

extern "C" void kernel_launch(void* const* d_in, const int* in_sizes, int n_in,
                              void* d_out, int out_size, void* d_ws, size_t ws_size,
                              hipStream_t stream) {
    const float* x = (const float*)d_in[0];      // [64,64,64,64]
    const float* z = (const float*)d_in[1];      // [64,4096,3]
    float* counts = (float*)d_out;               // output 0: [64,64,64,64]
    float* rmask  = (float*)d_out + NCOUNT;      // output 1: [64,64,64,64]

    // 1) zero the counts region (d_out is poisoned; must re-zero every call)
    zero_counts_kernel<<<16384, 256, 0, stream>>>((float4*)counts);
    // 2) histogram: one thread per point (64*4096 = 262144)
    hist_kernel<<<1024, 256, 0, stream>>>(z, counts);
    // 3) suffix-sum (triangular WMMA) + mask-multiply, in place on counts
    scan_mask_kernel<<<2048, 256, 0, stream>>>(x, counts, rmask);
}